// NautilusFusedKernel_88596585382518
// MI455X (gfx1250) — compile-verified
//
#include <hip/hip_runtime.h>
#include <math.h>

// ---------------------------------------------------------------------------
// NautilusLUT fused kernel for gfx1250 (MI455X, CDNA5, wave32).
//
// Rotation layers == fixed 128x128 matrix M (K-major packed in groups of 4 as
// P[g][col][k0..k3]); rows transformed with V_WMMA_F32_16X16X4_F32.
// P is staged global->LDS via GLOBAL_LOAD_ASYNC_TO_LDS_B128 (ASYNCcnt path),
// then B fragments are single ds_load_b64s.
// Memory-bound: ~240 MiB HBM traffic -> ~10.6 us floor @ 23.3 TB/s.
// ---------------------------------------------------------------------------

typedef __attribute__((ext_vector_type(2))) float v2f;
typedef __attribute__((ext_vector_type(8))) float v8f;
typedef int v4i __attribute__((__vector_size__(16)));

#define DIM    128
#define TILE   60
#define LEVELS 8

static __device__ __forceinline__ v8f wmma_f32_16x16x4(v2f a, v2f b, v8f c) {
  // 8 args: (neg_a, A, neg_b, B, c_mod, C, reuse_a, reuse_b)
  return __builtin_amdgcn_wmma_f32_16x16x4_f32(false, a, false, b, (short)0, c,
                                               false, false);
}

// ------------------------- Kernel 1: build packed M ------------------------
// Thread k computes row k of M (rotation applied to basis vector e_k), so
// out_row = in_row * M. Stored packed K-major in groups of 4:
//   P[((k>>2)*DIM + j)*4 + (k&3)] = M[k][j]
__global__ void build_rot_matrix(float* __restrict__ P) {
  const int k = threadIdx.x;  // 0..127
  float v[DIM];
#pragma unroll
  for (int j = 0; j < DIM; ++j) v[j] = 0.0f;
  v[k] = 1.0f;

  const double PHI = 1.61803398874989484820458683436;
  const double GA  = 2.0 * 3.14159265358979323846 / (PHI * PHI);

  // Layer 1: pairs (2m, 2m+1), angle GA*(m+1)
  for (int m = 0; m < DIM / 2; ++m) {
    const int i = 2 * m, j = 2 * m + 1;
    const double ang = GA * (double)(m + 1);
    const float c = (float)cos(ang), s = (float)sin(ang);
    const float a = v[i], b = v[j];
    v[i] = a * c - b * s;
    v[j] = a * s + b * c;
  }
  // Layer 2: pairs (2m+1, 2m+2), angle GA*(m+1)*PHI
  for (int m = 0; m < (DIM - 1) / 2; ++m) {
    const int i = 2 * m + 1, j = 2 * m + 2;
    const double ang = GA * (double)(m + 1) * PHI;
    const float c = (float)cos(ang), s = (float)sin(ang);
    const float a = v[i], b = v[j];
    v[i] = a * c - b * s;
    v[j] = a * s + b * c;
  }
  // Layer 3: stride 32; surviving pairs (kk, kk+32), kk in 0..31 and 64..95
  for (int m = 0; m < 64; ++m) {
    const int kk = (m < 32) ? m : (m + 32);
    const int i = kk, j = kk + 32;
    const double ang = GA * (double)(kk + 1) * PHI * PHI;
    const float c = (float)cos(ang), s = (float)sin(ang);
    const float a = v[i], b = v[j];
    v[i] = a * c - b * s;
    v[j] = a * s + b * c;
  }

  const int g  = k >> 2;
  const int kk = k & 3;
#pragma unroll
  for (int j = 0; j < DIM; ++j) P[((g * DIM + j) << 2) + kk] = v[j];
}

// ------------------------- Kernel 2: fused main ----------------------------
// One block per 60-row tile (padded to 64 rows). 256 threads = 8 waves;
// wave w: row-tile (w%4), col-tiles 4*(w/4)..+3.
// LDS (64 KB total, regions time-multiplexed):
//   phase 1: P packed matrix, 16384 floats (all 64 KB)
//   phase 2: t rotated tile 64*128 floats (32 KB) + mn/mx at +32 KB
__global__ __launch_bounds__(256) void nautilus_fused(
    const float* __restrict__ x, const float* __restrict__ Pg,
    float* __restrict__ out, int n) {
  __shared__ __align__(16) float smem[DIM * DIM];  // 64 KB

  const int tile = blockIdx.x;
  const int tid  = threadIdx.x;
  const int wave = tid >> 5;
  const int lane = tid & 31;
  const int half = lane >> 4;   // 0: lanes 0-15, 1: lanes 16-31
  const int ln   = lane & 15;

  const int rt = wave & 3;            // row tile index 0..3
  const int cb = (wave >> 2) * 4;     // first of 4 col-tiles (0 or 4)

  // ---- Stage packed M into LDS (async-to-LDS path when available) ----
  {
    const float4* __restrict__ src = (const float4*)Pg;
    float4* dst = (float4*)smem;
#if __has_builtin(__builtin_amdgcn_global_load_async_to_lds_b128)
    typedef __attribute__((address_space(1))) v4i* gv4i_p;
    typedef __attribute__((address_space(3))) v4i* lv4i_p;
#pragma unroll
    for (int it = 0; it < 16; ++it) {
      const int i = tid + it * 256;  // 4096 float4s
      __builtin_amdgcn_global_load_async_to_lds_b128(
          (gv4i_p)(src + i), (lv4i_p)(dst + i), 0, 0);
    }
#if __has_builtin(__builtin_amdgcn_s_wait_asynccnt)
    __builtin_amdgcn_s_wait_asynccnt(0);
#else
    asm volatile("s_wait_asynccnt 0x0" ::: "memory");
#endif
#else
#pragma unroll
    for (int it = 0; it < 16; ++it) {
      const int i = tid + it * 256;
      dst[i] = src[i];
    }
#endif
  }
  __syncthreads();

  // ---- GEMM: rows(16) x M via v_wmma_f32_16x16x4_f32 ----
  // A 16x4 f32 layout: lanes 0-15 hold K=0,1; lanes 16-31 hold K=2,3.
  int arow = tile * TILE + rt * 16 + ln;
  if (arow > n - 1) arow = n - 1;     // clamp padded rows of last tile
  const float* __restrict__ xrow = x + (size_t)arow * DIM;

  const int c0 = (cb + 0) * 16 + ln;
  const int c1 = (cb + 1) * 16 + ln;
  const int c2 = (cb + 2) * 16 + ln;
  const int c3 = (cb + 3) * 16 + ln;

  v8f acc0 = {}, acc1 = {}, acc2 = {}, acc3 = {};

#pragma unroll 4
  for (int g = 0; g < 32; ++g) {      // K groups of 4
    const v2f a = *(const v2f*)(xrow + 4 * g + 2 * half);
    const float* Pb = smem + ((g * DIM) << 2) + 2 * half;
    const v2f b0 = *(const v2f*)(Pb + (c0 << 2));
    const v2f b1 = *(const v2f*)(Pb + (c1 << 2));
    const v2f b2 = *(const v2f*)(Pb + (c2 << 2));
    const v2f b3 = *(const v2f*)(Pb + (c3 << 2));
    acc0 = wmma_f32_16x16x4(a, b0, acc0);
    acc1 = wmma_f32_16x16x4(a, b1, acc1);
    acc2 = wmma_f32_16x16x4(a, b2, acc2);
    acc3 = wmma_f32_16x16x4(a, b3, acc3);
  }
  __syncthreads();  // P region dead; reuse LDS below

  float* t   = smem;              // 64*128 floats (32 KB)
  float* mnv = smem + 64 * DIM;   // 128 floats at +32 KB
  float* mxv = mnv + DIM;

  // D layout (16x16 f32 C/D): VGPR v -> row (v + 8*half), col = lane%16.
#pragma unroll
  for (int v = 0; v < 8; ++v) {
    const int r = rt * 16 + v + 8 * half;
    t[r * DIM + c0] = acc0[v];
    t[r * DIM + c1] = acc1[v];
    t[r * DIM + c2] = acc2[v];
    t[r * DIM + c3] = acc3[v];
  }
  __syncthreads();

  // ---- Polar conversion in place: 60 rows * 64 pairs -> 15 per thread ----
#pragma unroll
  for (int it = 0; it < 15; ++it) {
    const int idx = tid + it * 256;
    const int r = idx >> 6;
    const int p = idx & 63;
    v2f ab = *(v2f*)&t[r * DIM + 2 * p];
    v2f pol;
    pol.x = sqrtf(ab.x * ab.x + ab.y * ab.y);
    pol.y = atan2f(ab.y, ab.x);
    *(v2f*)&t[r * DIM + 2 * p] = pol;
  }
  __syncthreads();

  // ---- Column min/max over the 60 valid rows ----
  if (tid < DIM) {
    float lo = t[tid], hi = t[tid];
    for (int r = 1; r < TILE; ++r) {
      const float p = t[r * DIM + tid];
      lo = fminf(lo, p);
      hi = fmaxf(hi, p);
    }
    mnv[tid] = lo;
    mxv[tid] = hi;
  }
  __syncthreads();

  // ---- Quantize / dequant / sign-correct; vectorized b64 stores ----
  const float lm1     = (float)(LEVELS - 1);
  const float inv_lm1 = 1.0f / lm1;
#pragma unroll
  for (int it = 0; it < 15; ++it) {
    const int idx = tid + it * 256;   // 0..3839 pair-slots
    const int r  = idx >> 6;
    const int cc = (idx & 63) * 2;
    const v2f p = *(const v2f*)&t[r * DIM + cc];
    v2f res;
#pragma unroll
    for (int e = 0; e < 2; ++e) {
      const float lo = mnv[cc + e];
      const float range = fmaxf(mxv[cc + e] - lo, 1e-8f);
      const float norm = (p[e] - lo) / range;
      float q = rintf(norm * lm1);            // RNE, matches jnp.round
      q = fminf(fmaxf(q, 0.0f), lm1);
      const float dq = q * inv_lm1 * range + lo;
      // corrected = dq + sign(err)*|err|*0.5 == dq + 0.5*(p - dq)
      res[e] = dq + 0.5f * (p[e] - dq);
    }
    *(v2f*)&out[((size_t)tile * TILE + r) * DIM + cc] = res;
  }
}

// ---------------------------------------------------------------------------
extern "C" void kernel_launch(void* const* d_in, const int* in_sizes, int n_in,
                              void* d_out, int out_size, void* d_ws,
                              size_t ws_size, hipStream_t stream) {
  const float* x = (const float*)d_in[0];
  float* out = (float*)d_out;
  float* P = (float*)d_ws;  // 128*128*4 = 64 KB scratch (packed M)

  const int n = in_sizes[0] / DIM;   // rows (245760)
  const int tiles = n / TILE;        // 4096

  build_rot_matrix<<<1, DIM, 0, stream>>>(P);
  nautilus_fused<<<tiles, 256, 0, stream>>>(x, P, out, n);
}